// TransitionDown_13975823581709
// MI455X (gfx1250) — compile-verified
//
#include <hip/hip_runtime.h>
#include <hip/hip_bf16.h>

// ---------------- problem constants ----------------
#define MPTS   60000      // M = N/STRIDE
#define K3C    40         // neighbors (branch3)
#define CIN    32
#define DIN    35         // 3 + CIN
#define DPAD   36         // K padded to multiple of 4
#define OC     64         // C_OUT
#define EPSBN  1e-5f

// d_out layout (floats): n_p [M*3] | y [M*64] | n_o [1 int bitcast]
// workspace layout (floats):
//   0     : gstats   [3*64*2]   branch sum/sumsq
//   384   : gstats2  [64*2]     final sum/sumsq
//   512   : ab_a     [192]      branch BN scale
//   704   : ab_c     [192]      branch BN shift
//   896   : fa       [64]       final BN scale
//   960   : fc       [64]       final BN shift
//   1024  : hmax     [M*192]    raw pre-BN branch maxima
//   1024+M*192 : ypre [M*64]
// total ~61.5 MB

typedef float v2f __attribute__((ext_vector_type(2)));
typedef float v8f __attribute__((ext_vector_type(8)));

// Order-preserving bijection IEEE-float -> uint32: float compare == unsigned compare.
// Branchless => single ds_max_u32, no EXEC manipulation.
__device__ __forceinline__ unsigned int fkey_encode(float v) {
  int b = __float_as_int(v);
  return (unsigned int)(b ^ ((b >> 31) | 0x80000000));
}
__device__ __forceinline__ float fkey_decode(unsigned int k) {
  int mask = ((int)k < 0) ? 0x80000000 : 0xFFFFFFFF;  // MSB set => original was >= +0
  return __int_as_float((int)(k ^ (unsigned int)mask));
}

// ---------------- kernel 0: zero stats ----------------
__global__ void k_zero(float* ws) {
  int i = threadIdx.x;
  for (; i < 512; i += 256) ws[i] = 0.0f;
}

// ---------------- kernel 1: gather + 3 branch GEMMs (fp32 WMMA) + max + stats ----------------
#define TM1 8   // centroids per block
__global__ __launch_bounds__(256) void k_branches(
    const float* __restrict__ p, const float* __restrict__ x,
    const int* __restrict__ fps_idx, const int* __restrict__ knn,
    const float* __restrict__ w1, const float* __restrict__ w2, const float* __restrict__ w3,
    float* __restrict__ ws, float* __restrict__ out_np) {

  __shared__ float        feat[TM1 * K3C][DPAD];  // 46080 B (cols: 0..31 = x, 32..34 = rel, 35 = 0)
  __shared__ float        wlds[DPAD][OC];         //  9216 B (row-permuted weights, row 35 = 0)
  __shared__ unsigned int hmaxl[3][TM1][OC];      //  6144 B (encoded float keys)
  __shared__ float        statl[3][OC][2];        //  1536 B
  __shared__ float        cent[TM1][4];           //   128 B

  const int tid  = threadIdx.x;
  const int lane = tid & 31;
  const int wave = tid >> 5;
  const int m0   = blockIdx.x * TM1;

  float* gstats = ws;                  // 384 floats
  float* hmax_g = ws + 1024;           // [M][192]

  // centroids + LDS init
  if (tid < TM1 * 3) {
    int ml = tid / 3, j = tid % 3;
    cent[ml][j] = p[(size_t)fps_idx[m0 + ml] * 3 + j];
  }
  for (int i = tid; i < 3 * TM1 * OC; i += 256) ((unsigned int*)hmaxl)[i] = 0u; // < key(any float)
  for (int i = tid; i < 3 * OC * 2;  i += 256) ((float*)statl)[i] = 0.0f;
  __syncthreads();

  // gather neighbor features into LDS (aligned float4 traffic)
  for (int r = tid; r < TM1 * K3C; r += 256) {
    int ml = r / K3C, kk = r % K3C;
    int idx = knn[(size_t)(m0 + ml) * K3C + kk];
    const float4* xr = (const float4*)(x + (size_t)idx * CIN);
    float4* fr = (float4*)feat[r];
#pragma unroll
    for (int j = 0; j < 8; ++j) fr[j] = xr[j];
    float4 rel;
    rel.x = p[(size_t)idx * 3 + 0] - cent[ml][0];
    rel.y = p[(size_t)idx * 3 + 1] - cent[ml][1];
    rel.z = p[(size_t)idx * 3 + 2] - cent[ml][2];
    rel.w = 0.0f;                                  // K-pad column 35
    fr[8] = rel;
  }

  const float* WS3[3] = {w1, w2, w3};
  const int    KS[3]  = {10, 20, 40};

  for (int s = 0; s < 3; ++s) {
    __syncthreads();
    // load weights with matching row permutation (x rows first, rel rows, zero pad)
    for (int i = tid; i < DPAD * OC; i += 256) {
      int k = i / OC, n = i % OC;
      float v = 0.0f;
      if (k < CIN)      v = WS3[s][(size_t)(k + 3) * OC + n];
      else if (k < DIN) v = WS3[s][(size_t)(k - CIN) * OC + n];
      wlds[k][n] = v;
    }
    __syncthreads();

    const int ks = KS[s];
    const int nTiles = (TM1 * ks / 16) * 4;       // row-tiles x 4 col-tiles
    for (int t = wave; t < nTiles; t += 8) {      // wave-uniform -> EXEC all ones at WMMA
      const int rt = t >> 2, ct = t & 3;
      const int row0 = rt * 16;
      const int myRow = row0 + (lane & 15);
      const int featRow = (myRow / ks) * K3C + (myRow % ks);
      const float* arow = feat[featRow];
      const int khalf = lane >> 4;
      const int col = ct * 16 + (lane & 15);

      v8f acc = {};
#pragma unroll
      for (int k0 = 0; k0 < DPAD; k0 += 4) {
        int ka = k0 + 2 * khalf;
        v2f a = *(const v2f*)(arow + ka);         // ds_load_b64 (8B aligned)
        v2f b; b.x = wlds[ka][col]; b.y = wlds[ka + 1][col];
        acc = __builtin_amdgcn_wmma_f32_16x16x4_f32(
            false, a, false, b, (short)0, acc, false, false);
      }

      const int rbase = row0 + khalf * 8;
      float sum = 0.0f, sq = 0.0f;
#pragma unroll
      for (int i = 0; i < 8; ++i) {
        float v = acc[i];
        sum += v; sq += v * v;
        int ml = (rbase + i) / ks;
        atomicMax(&hmaxl[s][ml][col], fkey_encode(v));   // branchless ds_max_u32
      }
      atomicAdd(&statl[s][col][0], sum);
      atomicAdd(&statl[s][col][1], sq);
    }
  }
  __syncthreads();

  // flush stats + maxima (decode keys) + centroids
  for (int i = tid; i < 3 * OC * 2; i += 256) atomicAdd(&gstats[i], ((const float*)statl)[i]);
  for (int i = tid; i < 3 * TM1 * OC; i += 256) {
    int s = i / (TM1 * OC), ml = (i / OC) % TM1, c = i % OC;
    hmax_g[(size_t)(m0 + ml) * 192 + s * 64 + c] =
        fkey_decode(((const unsigned int*)hmaxl)[i]);
  }
  if (tid < TM1 * 3) out_np[(size_t)(m0 + tid / 3) * 3 + tid % 3] = cent[tid / 3][tid % 3];
}

// ---------------- kernel 2: finalize branch BN affines ----------------
__global__ void k_fin_branch(float* ws, const float* g1, const float* g2, const float* g3,
                             const float* be1, const float* be2, const float* be3) {
  int ch = threadIdx.x;
  if (ch >= 192) return;
  int s = ch >> 6, c = ch & 63;
  const float* G[3]  = {g1, g2, g3};
  const float* BE[3] = {be1, be2, be3};
  float cnt = (float)MPTS * (float)(10 << s);
  float sum = ws[s * 128 + c * 2 + 0];
  float sq  = ws[s * 128 + c * 2 + 1];
  float mu  = sum / cnt;
  float var = sq / cnt - mu * mu;
  float a   = G[s][c] * rsqrtf(var + EPSBN);
  ws[512 + ch] = a;
  ws[704 + ch] = BE[s][c] - mu * a;
}

// ---------------- kernel 3: affine+ReLU then [M,192]@[192,64]+b (fp32 WMMA) + stats ----------------
#define TM3 32
__global__ __launch_bounds__(256) void k_final_gemm(
    float* __restrict__ ws, const float* __restrict__ w, const float* __restrict__ bvec) {

  __shared__ float catl[TM3][192];     // 24576 B
  __shared__ float statl[OC][2];       //   512 B

  const int tid  = threadIdx.x;
  const int lane = tid & 31;
  const int wave = tid >> 5;
  const int m0   = blockIdx.x * TM3;

  const float* ab_a   = ws + 512;
  const float* ab_c   = ws + 704;
  const float* hmax_g = ws + 1024;
  float*       gst2   = ws + 384;
  float*       ypre   = ws + 1024 + (size_t)MPTS * 192;

  for (int i = tid; i < TM3 * 192; i += 256) {
    int r = i / 192, ch = i % 192;
    float v = ab_a[ch] * hmax_g[(size_t)(m0 + r) * 192 + ch] + ab_c[ch];
    catl[r][ch] = v > 0.0f ? v : 0.0f;
  }
  for (int i = tid; i < OC * 2; i += 256) ((float*)statl)[i] = 0.0f;
  __syncthreads();

  // one 16x16 tile per wave: 2 row-tiles x 4 col-tiles, K=192
  const int rt = wave >> 2, ct = wave & 3;
  const int row0 = rt * 16;
  const float* arow = catl[row0 + (lane & 15)];
  const int khalf = lane >> 4;
  const int col = ct * 16 + (lane & 15);

  v8f acc = {};
#pragma unroll 4
  for (int k0 = 0; k0 < 192; k0 += 4) {
    int ka = k0 + 2 * khalf;
    v2f a = *(const v2f*)(arow + ka);
    v2f b; b.x = w[(size_t)ka * OC + col]; b.y = w[(size_t)(ka + 1) * OC + col];
    acc = __builtin_amdgcn_wmma_f32_16x16x4_f32(
        false, a, false, b, (short)0, acc, false, false);
  }

  const float bias = bvec[col];
  const int rbase = m0 + row0 + khalf * 8;
  float sum = 0.0f, sq = 0.0f;
#pragma unroll
  for (int i = 0; i < 8; ++i) {
    float v = acc[i] + bias;
    sum += v; sq += v * v;
    ypre[(size_t)(rbase + i) * OC + col] = v;
  }
  atomicAdd(&statl[col][0], sum);
  atomicAdd(&statl[col][1], sq);
  __syncthreads();
  for (int i = tid; i < OC * 2; i += 256) atomicAdd(&gst2[i], ((const float*)statl)[i]);
}

// ---------------- kernel 4: finalize final BN ----------------
__global__ void k_fin_final(float* ws, const float* g, const float* be) {
  int c = threadIdx.x;
  if (c >= 64) return;
  float cnt = (float)MPTS;
  float sum = ws[384 + c * 2 + 0];
  float sq  = ws[384 + c * 2 + 1];
  float mu  = sum / cnt;
  float var = sq / cnt - mu * mu;
  float a   = g[c] * rsqrtf(var + EPSBN);
  ws[896 + c] = a;
  ws[960 + c] = be[c] - mu * a;
}

// ---------------- kernel 5: apply final BN+ReLU, write y and n_o ----------------
__global__ void k_apply(const float* __restrict__ ws, float* __restrict__ out_y,
                        int* __restrict__ out_no) {
  size_t i = (size_t)blockIdx.x * 256 + threadIdx.x;
  const float* fa   = ws + 896;
  const float* fc   = ws + 960;
  const float* ypre = ws + 1024 + (size_t)MPTS * 192;
  if (i < (size_t)MPTS * OC) {
    int c = (int)(i & 63);
    float v = fa[c] * ypre[i] + fc[c];
    out_y[i] = v > 0.0f ? v : 0.0f;
  }
  if (i == 0) *out_no = MPTS;
}

// ---------------- host launcher ----------------
extern "C" void kernel_launch(void* const* d_in, const int* in_sizes, int n_in,
                              void* d_out, int out_size, void* d_ws, size_t ws_size,
                              hipStream_t stream) {
  const float* p       = (const float*)d_in[0];
  const float* x       = (const float*)d_in[1];
  // d_in[2] = o (unused)
  const int*   fps_idx = (const int*)d_in[3];
  const int*   knn_idx = (const int*)d_in[4];
  const float* w1      = (const float*)d_in[5];
  const float* w2      = (const float*)d_in[6];
  const float* w3      = (const float*)d_in[7];
  const float* w       = (const float*)d_in[8];
  const float* b       = (const float*)d_in[9];
  const float* g1      = (const float*)d_in[10];
  const float* g2      = (const float*)d_in[11];
  const float* g3      = (const float*)d_in[12];
  const float* g       = (const float*)d_in[13];
  const float* be1     = (const float*)d_in[14];
  const float* be2     = (const float*)d_in[15];
  const float* be3     = (const float*)d_in[16];
  const float* be      = (const float*)d_in[17];

  float* out    = (float*)d_out;
  float* out_np = out;
  float* out_y  = out + (size_t)MPTS * 3;
  int*   out_no = (int*)(out + (size_t)MPTS * 3 + (size_t)MPTS * OC);
  float* ws     = (float*)d_ws;

  k_zero<<<1, 256, 0, stream>>>(ws);
  k_branches<<<MPTS / TM1, 256, 0, stream>>>(p, x, fps_idx, knn_idx, w1, w2, w3, ws, out_np);
  k_fin_branch<<<1, 256, 0, stream>>>(ws, g1, g2, g3, be1, be2, be3);
  k_final_gemm<<<MPTS / TM3, 256, 0, stream>>>(ws, w, b);
  k_fin_final<<<1, 64, 0, stream>>>(ws, g, be);
  k_apply<<<(MPTS * OC + 255) / 256, 256, 0, stream>>>(ws, out_y, out_no);
}